// Mean_28759101014396
// MI455X (gfx1250) — compile-verified
//
#include <hip/hip_runtime.h>
#include <hip/hip_bf16.h>

#define B_    32
#define T_    2048
#define DIN   1024
#define DOUT  256

typedef __attribute__((ext_vector_type(2))) float v2f;
typedef __attribute__((ext_vector_type(4))) float v4f;
typedef __attribute__((ext_vector_type(8))) float v8f;

// ---------------- Kernel 1: per-sample valid length ----------------
// length = (first index with att_mask<0) + 1, else T  (mirrors reference)
__global__ void len_kernel(const float* __restrict__ att,
                           int* __restrict__ len, float* __restrict__ invlen) {
    int b = blockIdx.x;
    const float* m = att + (size_t)b * T_;
    int local = T_;
    for (int t = threadIdx.x; t < T_; t += blockDim.x)
        if (m[t] < 0.0f && t < local) local = t;
    __shared__ int red[256];
    red[threadIdx.x] = local;
    __syncthreads();
    for (int s = 128; s > 0; s >>= 1) {
        if (threadIdx.x < (unsigned)s)
            red[threadIdx.x] = min(red[threadIdx.x], red[threadIdx.x + s]);
        __syncthreads();
    }
    if (threadIdx.x == 0) {
        int first = red[0];
        int L = (first < T_) ? (first + 1) : T_;
        len[b] = L;
        invlen[b] = 1.0f / (float)L;
    }
}

// ---------------- Kernel 2: streaming ReLU row-sum (per chunk) ----------------
// partial[b][c][d] = sum over rows t in chunk c (t < L_b) of relu(feature[b][t][d])
__global__ void partial_kernel(const float* __restrict__ feature,
                               const int* __restrict__ len,
                               float* __restrict__ partial, int nchunk) {
    int c = blockIdx.x;
    int b = blockIdx.y;
    int rows = T_ / nchunk;
    int t0 = c * rows;
    int L  = len[b];
    int t1 = t0 + rows; if (t1 > L) t1 = L;

    int col = threadIdx.x * 4;                       // 256 thr * 4 = 1024 cols
    const float* base = feature + (size_t)b * T_ * DIN + col;

    v4f acc = {};
    for (int t = t0; t < t1; ++t) {
        v4f v = __builtin_nontemporal_load((v4f*)(base + (size_t)t * DIN));
        acc.x += fmaxf(v.x, 0.0f);
        acc.y += fmaxf(v.y, 0.0f);
        acc.z += fmaxf(v.z, 0.0f);
        acc.w += fmaxf(v.w, 0.0f);
    }
    *(v4f*)(partial + ((size_t)(b * nchunk + c)) * DIN + col) = acc;
}

// ---------------- Kernel 3: fold chunk partials into s[32][1024] ----------------
__global__ void reduce_kernel(const float* __restrict__ partial,
                              float* __restrict__ s, int nchunk) {
    int b = blockIdx.x;
    int col = threadIdx.x * 4;
    v4f acc = {};
    for (int c = 0; c < nchunk; ++c) {
        v4f v = *(const v4f*)(partial + ((size_t)(b * nchunk + c)) * DIN + col);
        acc.x += v.x; acc.y += v.y; acc.z += v.z; acc.w += v.w;
    }
    *(v4f*)(s + (size_t)b * DIN + col) = acc;
}

// ---------------- Kernel 4: [32x1024] x [1024x256] f32 GEMM via WMMA ----------------
// out[m][o] = dot(s[m,:], W[o,:]) * invlen[m] + bias[o]
// One wave per 16x16 output tile; 32 waves in one block cover 2x16 tiles.
__global__ __launch_bounds__(1024)
void wmma_gemm(const float* __restrict__ s, const float* __restrict__ W,
               const float* __restrict__ bias, const float* __restrict__ invlen,
               float* __restrict__ out) {
    int wid  = threadIdx.x >> 5;     // 0..31
    int lane = threadIdx.x & 31;
    int mtile = wid >> 4;            // 0..1  (M = 32)
    int ntile = wid & 15;            // 0..15 (N = 256)
    int half = lane >> 4;            // 0 or 1
    int lr   = lane & 15;

    // A (16x4 f32): lane lr holds row M=lr; VGPR0/1 hold K = 2*half + {0,1}
    const float* sA = s + (size_t)(mtile * 16 + lr) * DIN + 2 * half;
    // B (4x16 f32): lane lr holds col N=lr; VGPR0/1 hold K = 2*half + {0,1}
    const float* wB = W + (size_t)(ntile * 16 + lr) * DIN + 2 * half;

    v8f acc = {};
#pragma unroll 4
    for (int k = 0; k < DIN; k += 4) {
        v2f a  = *(const v2f*)(sA + k);
        v2f bb = *(const v2f*)(wB + k);
        acc = __builtin_amdgcn_wmma_f32_16x16x4_f32(
            /*neg_a=*/false, a, /*neg_b=*/false, bb,
            /*c_mod=*/(short)0, acc, /*reuse_a=*/false, /*reuse_b=*/false);
    }

    int o = ntile * 16 + lr;
    float bo = bias[o];
#pragma unroll
    for (int r = 0; r < 8; ++r) {
        int m = mtile * 16 + r + 8 * half;       // D layout: M = r + 8*(lane>=16)
        out[(size_t)m * DOUT + o] = acc[r] * invlen[m] + bo;
    }
}

extern "C" void kernel_launch(void* const* d_in, const int* in_sizes, int n_in,
                              void* d_out, int out_size, void* d_ws, size_t ws_size,
                              hipStream_t stream) {
    const float* feature = (const float*)d_in[0];   // [32,2048,1024]
    const float* att     = (const float*)d_in[1];   // [32,2048]
    const float* W       = (const float*)d_in[2];   // [256,1024]
    const float* bias    = (const float*)d_in[3];   // [256]
    float* out = (float*)d_out;                     // [32,256]

    char* ws = (char*)d_ws;
    float* invlen  = (float*)ws;                        // 32 f32
    int*   len     = (int*)(ws + 128);                  // 32 i32
    float* s       = (float*)(ws + 256);                // 32*1024 f32 = 128 KB
    size_t fixed   = 256 + (size_t)B_ * DIN * sizeof(float);
    float* partial = (float*)(ws + fixed);

    size_t per_chunk = (size_t)B_ * DIN * sizeof(float);   // 128 KB per chunk
    int nchunk = 32;
    while (nchunk > 1 && fixed + (size_t)nchunk * per_chunk > ws_size) nchunk >>= 1;

    len_kernel<<<dim3(B_), dim3(256), 0, stream>>>(att, len, invlen);
    partial_kernel<<<dim3(nchunk, B_), dim3(256), 0, stream>>>(feature, len, partial, nchunk);
    reduce_kernel<<<dim3(B_), dim3(256), 0, stream>>>(partial, s, nchunk);
    wmma_gemm<<<dim3(1), dim3(1024), 0, stream>>>(s, W, bias, invlen, out);
}